// Multivector_34892314313417
// MI455X (gfx1250) — compile-verified
//
#include <hip/hip_runtime.h>
#include <hip/hip_bf16.h>

typedef float v2f __attribute__((ext_vector_type(2)));
typedef float v8f __attribute__((ext_vector_type(8)));

// ---------------------------------------------------------------------------
// Setup: build the combined 16x16 sandwich matrix C_b per batch into d_ws.
// out = C_b * x  with  C_b = B2_b @ A_b  (see analysis).  16 blocks x 256 thr.
// For fixed column, each (k,col) target has exactly ONE writer -> no atomics.
// ---------------------------------------------------------------------------
__global__ void build_cmat_kernel(const float* __restrict__ motor,
                                  float* __restrict__ cmat) {
    __shared__ float A[16][16];   // A[k][j]  (acts on x)
    __shared__ float B2[16][16];  // B2[k][l] (acts on tmp)

    const int b   = blockIdx.x;
    const int t   = threadIdx.x;
    const int row = t >> 4;   // i (for A) / l (for B2) / output row
    const int col = t & 15;   // j                         / output col

    // basis masks sorted by (popcount, value) and the inverse permutation
    const int bits_of[16] = {0,1,2,4,8,3,5,6,9,10,12,7,11,13,14,15};
    const int idx_of[16]  = {0,1,2,5,3,6,7,11,4,8,9,12,10,13,14,15};
    const float rev_sign[16] = {1,1,1,1,1,-1,-1,-1,-1,-1,-1,-1,-1,-1,-1,1};

    // --- A[k][j] = T[i,j,k] * m_i   (k = idx(bi^bj), unique i per (k,j)) ---
    {
        const int bi = bits_of[row], bj = bits_of[col];
        const int k  = idx_of[bi ^ bj];
        int swaps = 0;
#pragma unroll
        for (int kk = 0; kk < 4; ++kk)
            if ((bi >> kk) & 1) swaps += __popc(bj & ((1 << kk) - 1));
        const float coeff = (swaps & 1) ? -1.0f : 1.0f;
        A[k][col] = (bi & bj & 8) ? 0.0f : coeff * motor[b * 16 + row];
    }
    // --- B2[k][l] = T[l,j,k] * mrev_j  (unique j per (k,l)) ---
    {
        const int bl = bits_of[row], bj = bits_of[col];
        const int k  = idx_of[bl ^ bj];
        int swaps = 0;
#pragma unroll
        for (int kk = 0; kk < 4; ++kk)
            if ((bl >> kk) & 1) swaps += __popc(bj & ((1 << kk) - 1));
        const float coeff = (swaps & 1) ? -1.0f : 1.0f;
        B2[k][row] = (bl & bj & 8)
                         ? 0.0f
                         : coeff * (motor[b * 16 + col] * rev_sign[col]);
    }
    __syncthreads();

    // --- C = B2 @ A, stored row-major: C[row][col] at b*256 + row*16 + col ---
    float s = 0.0f;
#pragma unroll
    for (int k = 0; k < 16; ++k) s += B2[row][k] * A[k][col];
    cmat[b * 256 + row * 16 + col] = s;
}

// ---------------------------------------------------------------------------
// Streaming kernel: OUT^T(tile 16 vec x 16 comp) = X^T(16x16) * C^T(16x16)
// via 4x V_WMMA_F32_16X16X4_F32 (K=16 accumulated in 4 slices of 4).
// Grid: (blocksPerBatch, 16 batches), block = 256 threads = 8 wave32.
// Whole-block uniform control flow -> EXEC all ones for WMMA.
// ---------------------------------------------------------------------------
__global__ void __launch_bounds__(256)
pga_sandwich_kernel(const float* __restrict__ x,
                    const float* __restrict__ cmat,
                    float* __restrict__ out,
                    int tilesPerBatch) {
    const int b    = blockIdx.y;
    const int lane = threadIdx.x & 31;
    const int l16  = lane & 15;     // M row (A) / N col (B,D)
    const int half = lane >> 4;     // K-half selector

    const int wavesPerBatch = gridDim.x * (blockDim.x >> 5);
    const int waveId = blockIdx.x * (blockDim.x >> 5) + (threadIdx.x >> 5);

    const size_t batchOff = (size_t)b * (size_t)tilesPerBatch * 256u;
    const float* __restrict__ xb = x   + batchOff;
    float*       __restrict__ ob = out + batchOff;
    const float* __restrict__ cb = cmat + b * 256;

    // B operand (C^T K-slices) lives in VGPRs for the whole loop.
    // B slice j: element (K = 2*half + v, N = l16) = C[l16][4j + 2*half + v]
    v2f bm[4];
#pragma unroll
    for (int j = 0; j < 4; ++j)
        bm[j] = *(const v2f*)(cb + l16 * 16 + 4 * j + 2 * half);

    const int laneOffA = l16 * 16 + 2 * half;  // + 4*j per K-slice

    for (int t = waveId; t < tilesPerBatch; t += wavesPerBatch) {
        const float* __restrict__ xt = xb + (size_t)t * 256u;

        // A operand: lane L holds x[vec l16][4j + 2*half .. +1]  (b64 loads,
        // every element of the 1 KB tile loaded exactly once, NT to skip L2)
        v2f a[4];
#pragma unroll
        for (int j = 0; j < 4; ++j)
            a[j] = __builtin_nontemporal_load((const v2f*)(xt + laneOffA + 4 * j));

        v8f c = {0.0f, 0.0f, 0.0f, 0.0f, 0.0f, 0.0f, 0.0f, 0.0f};
#pragma unroll
        for (int j = 0; j < 4; ++j)
            c = __builtin_amdgcn_wmma_f32_16x16x4_f32(
                    /*neg_a=*/false, a[j], /*neg_b=*/false, bm[j],
                    /*c_mod=*/(short)0, c, /*reuse_a=*/false, /*reuse_b=*/false);

        // D VGPR r, lane L -> (vector r + 8*half, component l16)
        float* __restrict__ ot = ob + (size_t)t * 256u;
#pragma unroll
        for (int r = 0; r < 8; ++r)
            __builtin_nontemporal_store(c[r], ot + (r + 8 * half) * 16 + l16);
    }
}

extern "C" void kernel_launch(void* const* d_in, const int* in_sizes, int n_in,
                              void* d_out, int out_size, void* d_ws, size_t ws_size,
                              hipStream_t stream) {
    const float* x     = (const float*)d_in[0];   // (16, N, 16) f32
    const float* motor = (const float*)d_in[1];   // (16, 16)    f32
    float* out  = (float*)d_out;
    float* cmat = (float*)d_ws;                   // 16 * 256 floats = 16 KB

    const int nPerBatch     = in_sizes[0] / (16 * 16); // vectors per batch (262144)
    const int tilesPerBatch = nPerBatch / 16;          // 16-vector tiles (16384)

    build_cmat_kernel<<<16, 256, 0, stream>>>(motor, cmat);

    // 256 blocks/batch * 8 waves = 2048 waves per batch -> 8 tiles per wave.
    pga_sandwich_kernel<<<dim3(256, 16), 256, 0, stream>>>(x, cmat, out,
                                                           tilesPerBatch);
}